// SimpleEventGenerator_67310727463601
// MI455X (gfx1250) — compile-verified
//
#include <hip/hip_runtime.h>

// ---------------- problem constants ----------------
#define B_       32
#define CTX      128
#define NFRAMES  128
#define NSAMPLES 32768
#define NEVENTS  16
#define CH       256
#define WIN      512
#define HOP      256
#define NCOEFFS  257
#define TC       514      // TOTAL_COEFFS
#define KD       516      // DFT K padded to multiple of 4
#define MROWS    4096     // B_*NFRAMES

typedef __attribute__((ext_vector_type(2))) float v2f;
typedef __attribute__((ext_vector_type(8))) float v8f;

#define PI_F 3.14159265358979323846f

#define WMMA_F32(a, b, c) \
    __builtin_amdgcn_wmma_f32_16x16x4_f32(false, (a), false, (b), (short)0, (c), false, false)

// ---------------------------------------------------------------------------
// proj: x0[b,c] = sum_k param[b,k]*proj_w[c,k] + proj_b[c]     (32 x 256)
// ---------------------------------------------------------------------------
__global__ void proj_kernel(const float* __restrict__ param,
                            const float* __restrict__ pw,
                            const float* __restrict__ pb,
                            float* __restrict__ x0) {
    int tid = blockIdx.x * blockDim.x + threadIdx.x;
    if (tid >= B_ * CH) return;
    int b = tid / CH, c = tid % CH;
    const float* pr = param + b * CTX;
    const float* wr = pw + c * CTX;
    float acc = 0.f;
#pragma unroll 8
    for (int k = 0; k < CTX; ++k) acc += pr[k] * wr[k];
    x0[tid] = acc + pb[c];
}

// X[(b*128+f), c] = x0[b,c] + pos[f,c]
__global__ void prepx_kernel(const float* __restrict__ x0,
                             const float* __restrict__ pos,
                             float* __restrict__ X) {
    int tid = blockIdx.x * blockDim.x + threadIdx.x;      // MROWS*CH threads
    int c = tid % CH;
    int r = tid / CH;
    int f = r % NFRAMES;
    int b = r / NFRAMES;
    X[tid] = x0[b * CH + c] + pos[f * CH + c];
}

// ---------------------------------------------------------------------------
// WMMA f32 GEMM:  out[m,n] = act( sum_k A[m,k]*W[n,k] + bias[n] )
// A: M x K (lda), W: Nw x K row-major (ldb), out: M x Nw (ldo).
// One wave computes a 32x64 tile: 2 A fragments x 4 B fragments = 8 WMMAs
// per K-step of 4 (6 fragment loads / 8 WMMAs). Grid covers (M/32)*nstrips
// tiles, 8 waves/block. Full strips take an unmasked loop (wave-uniform
// branch, EXEC stays all-ones); only the edge strip pays clamp+mask cost.
// ---------------------------------------------------------------------------
__global__ __launch_bounds__(256)
void gemm_wmma(const float* __restrict__ A, const float* __restrict__ W,
               const float* __restrict__ bias, float* __restrict__ out,
               int M, int Nw, int K, int lda, int ldb, int ldo,
               int nstrips, int act) {
    int wave = threadIdx.x >> 5;
    int lane = threadIdx.x & 31;
    int tile = blockIdx.x * 8 + wave;
    int ntiles = (M >> 5) * nstrips;
    if (tile >= ntiles) return;                 // whole-wave uniform exit
    int mt = tile / nstrips;
    int ns = tile % nstrips;
    int m0 = mt << 5;
    int n0 = ns << 6;

    // A fragment addressing (ISA 32-bit A 16x4 layout):
    //   lane L: row m0+(L&15), K pair kk0 = 2*(L>>4)
    int rA  = m0 + (lane & 15);
    int kk0 = (lane >> 4) << 1;
    const float* Ap0 = A + (size_t)rA * lda + kk0;
    const float* Ap1 = Ap0 + (size_t)16 * lda;

    int rB0 = n0 + (lane & 15);
    bool full = (n0 + 64 <= Nw);                // wave-uniform

    v8f a00 = {}, a01 = {}, a02 = {}, a03 = {};
    v8f a10 = {}, a11 = {}, a12 = {}, a13 = {};

    if (full) {
        const float* Bp0 = W + (size_t)rB0 * ldb + kk0;
        const float* Bp1 = Bp0 + (size_t)16 * ldb;
        const float* Bp2 = Bp0 + (size_t)32 * ldb;
        const float* Bp3 = Bp0 + (size_t)48 * ldb;
        for (int k = 0; k < K; k += 4) {
            v2f fa0 = *(const v2f*)(Ap0 + k);
            v2f fa1 = *(const v2f*)(Ap1 + k);
            v2f fb0 = *(const v2f*)(Bp0 + k);
            v2f fb1 = *(const v2f*)(Bp1 + k);
            v2f fb2 = *(const v2f*)(Bp2 + k);
            v2f fb3 = *(const v2f*)(Bp3 + k);
            a00 = WMMA_F32(fa0, fb0, a00);
            a10 = WMMA_F32(fa1, fb0, a10);
            a01 = WMMA_F32(fa0, fb1, a01);
            a11 = WMMA_F32(fa1, fb1, a11);
            a02 = WMMA_F32(fa0, fb2, a02);
            a12 = WMMA_F32(fa1, fb2, a12);
            a03 = WMMA_F32(fa0, fb3, a03);
            a13 = WMMA_F32(fa1, fb3, a13);
        }
    } else {
        // edge strip: clamp rows, arithmetic zero-mask (no EXEC divergence)
        int rBn0 = rB0;      int rBc0 = rBn0 < Nw ? rBn0 : Nw - 1;  float mk0 = rBn0 < Nw ? 1.f : 0.f;
        int rBn1 = rB0 + 16; int rBc1 = rBn1 < Nw ? rBn1 : Nw - 1;  float mk1 = rBn1 < Nw ? 1.f : 0.f;
        int rBn2 = rB0 + 32; int rBc2 = rBn2 < Nw ? rBn2 : Nw - 1;  float mk2 = rBn2 < Nw ? 1.f : 0.f;
        int rBn3 = rB0 + 48; int rBc3 = rBn3 < Nw ? rBn3 : Nw - 1;  float mk3 = rBn3 < Nw ? 1.f : 0.f;
        const float* Bp0 = W + (size_t)rBc0 * ldb + kk0;
        const float* Bp1 = W + (size_t)rBc1 * ldb + kk0;
        const float* Bp2 = W + (size_t)rBc2 * ldb + kk0;
        const float* Bp3 = W + (size_t)rBc3 * ldb + kk0;
        for (int k = 0; k < K; k += 4) {
            v2f fa0 = *(const v2f*)(Ap0 + k);
            v2f fa1 = *(const v2f*)(Ap1 + k);
            v2f fb0 = *(const v2f*)(Bp0 + k) * mk0;
            v2f fb1 = *(const v2f*)(Bp1 + k) * mk1;
            v2f fb2 = *(const v2f*)(Bp2 + k) * mk2;
            v2f fb3 = *(const v2f*)(Bp3 + k) * mk3;
            a00 = WMMA_F32(fa0, fb0, a00);
            a10 = WMMA_F32(fa1, fb0, a10);
            a01 = WMMA_F32(fa0, fb1, a01);
            a11 = WMMA_F32(fa1, fb1, a11);
            a02 = WMMA_F32(fa0, fb2, a02);
            a12 = WMMA_F32(fa1, fb2, a12);
            a03 = WMMA_F32(fa0, fb3, a03);
            a13 = WMMA_F32(fa1, fb3, a13);
        }
    }

    // C/D layout: VGPR v, lanes 0-15: M=v, N=lane; lanes 16-31: M=8+v, N=lane-16
    int mrow = m0 + ((lane >> 4) << 3);
    int col0 = lane & 15;
#pragma unroll
    for (int v = 0; v < 8; ++v) {
        float v0[4] = { a00[v], a01[v], a02[v], a03[v] };
        float v1[4] = { a10[v], a11[v], a12[v], a13[v] };
#pragma unroll
        for (int j = 0; j < 4; ++j) {
            int col = n0 + (j << 4) + col0;
            if (col < Nw) {
                float bsum = bias ? bias[col] : 0.f;
                float x = v0[j] + bsum;
                if (act) x = x > 0.f ? x : 0.2f * x;
                out[(size_t)(mrow + v) * ldo + col] = x;
                float y = v1[j] + bsum;
                if (act) y = y > 0.f ? y : 0.2f * y;
                out[(size_t)(mrow + 16 + v) * ldo + col] = y;
            }
        }
    }
}

// ---------------------------------------------------------------------------
// Spectrum / cumsum kernel. One thread per (b, j) coefficient.
// A_dft[row, 2j] = |y_even| * cos(p),  A_dft[row, 2j+1] = |y_even| * sin(p)
// p = cumsum over frames of (1 + y_odd * noise) * (j/256) * pi
// ---------------------------------------------------------------------------
__global__ void spec_kernel(const float* __restrict__ Y,
                            const float* __restrict__ noise,
                            float* __restrict__ Adft) {
    int tid = blockIdx.x * blockDim.x + threadIdx.x;
    if (tid >= B_ * NCOEFFS) return;
    int b = tid / NCOEFFS, j = tid % NCOEFFS;
    float freq = ((float)j / 256.0f) * PI_F;
    float p = 0.f;
    for (int f = 0; f < NFRAMES; ++f) {
        int row = b * NFRAMES + f;
        float y0 = Y[(size_t)row * TC + 2 * j];
        float y1 = Y[(size_t)row * TC + 2 * j + 1];
        float nz = noise[(size_t)row * NCOEFFS + j];
        p += (1.0f + y1 * nz) * freq;
        float mag = fabsf(y0);
        float s, c;
        sincosf(p, &s, &c);
        Adft[(size_t)row * KD + 2 * j]     = mag * c;
        Adft[(size_t)row * KD + 2 * j + 1] = mag * s;
        if (j == 0) {   // zero the K-padding columns for every row
            Adft[(size_t)row * KD + 514] = 0.f;
            Adft[(size_t)row * KD + 515] = 0.f;
        }
    }
}

// ---------------------------------------------------------------------------
// DFT matrix builder, stored N x K (K-contiguous) for the GEMM's A*W^T form:
//   Wd[n, 2j]   =  c_j * cos(2*pi*j*n/512) * hann(n) / 512
//   Wd[n, 2j+1] = -c_j * sin(2*pi*j*n/512) * hann(n) / 512   (0 at j=0, 256)
// c_j = 1 for j in {0,256}, else 2. Imag of DC/Nyquist ignored per irfft.
// ---------------------------------------------------------------------------
__global__ void wdft_kernel(float* __restrict__ Wd) {
    int tid = blockIdx.x * blockDim.x + threadIdx.x;
    if (tid >= WIN * KD) return;
    int n = tid / KD, k = tid % KD;
    float val = 0.f;
    if (k < TC) {
        int j = k >> 1;
        int edge = (j == 0) | (j == 256);
        float cj = edge ? 1.0f : 2.0f;
        float ang = 2.0f * PI_F * (float)j * (float)n / 512.0f;
        float h = 0.5f * (1.0f - cosf(2.0f * PI_F * (float)n / 512.0f));
        if ((k & 1) == 0) val = cj * cosf(ang) * h * (1.0f / 512.0f);
        else              val = edge ? 0.f : -cj * sinf(ang) * h * (1.0f / 512.0f);
    }
    Wd[tid] = val;
}

// ---------------------------------------------------------------------------
// Overlap-add: each t in [0,32768) is covered by frames f0=t/256 and f0-1.
// ---------------------------------------------------------------------------
__global__ void ola_kernel(const float* __restrict__ frames,
                           float* __restrict__ sig) {
    int tid = blockIdx.x * blockDim.x + threadIdx.x;   // B_*NSAMPLES
    int b = tid >> 15;
    int t = tid & (NSAMPLES - 1);
    int f0 = t >> 8;
    int n0 = t & 255;
    float v = frames[(size_t)(b * NFRAMES + f0) * WIN + n0];
    if (f0 > 0) v += frames[(size_t)(b * NFRAMES + f0 - 1) * WIN + n0 + 256];
    sig[tid] = v;
}

// ---------------------------------------------------------------------------
// Sparse "FFT conv": out[b,e,t] = sum_k times[b,e,k] * sig[b, t - 256k]
// (linear conv: 2N-padded circular conv never wraps, truncated to N).
// Taps compacted into LDS once per block (typically a single nonzero).
// ---------------------------------------------------------------------------
__global__ __launch_bounds__(256)
void conv_kernel(const float* __restrict__ times,
                 const float* __restrict__ sig,
                 float* __restrict__ out) {
    __shared__ int   s_d[NFRAMES];
    __shared__ float s_a[NFRAMES];
    __shared__ int   s_cnt;
    int chunk = blockIdx.x & 15;          // 16 chunks of 2048 samples
    int e = (blockIdx.x >> 4) & 15;
    int b = blockIdx.x >> 8;
    if (threadIdx.x == 0) {
        int cnt = 0;
        const float* tr = times + (size_t)(b * NEVENTS + e) * NFRAMES;
        for (int k = 0; k < NFRAMES; ++k) {
            float a = tr[k];
            if (a != 0.f) { s_d[cnt] = k * HOP; s_a[cnt] = a; ++cnt; }
        }
        s_cnt = cnt;
    }
    __syncthreads();
    int cnt = s_cnt;
    const float* sg = sig + (size_t)b * NSAMPLES;
    float* op = out + (size_t)(b * NEVENTS + e) * NSAMPLES + chunk * 2048;
#pragma unroll
    for (int i = 0; i < 8; ++i) {
        int t = chunk * 2048 + i * 256 + (int)threadIdx.x;
        float acc = 0.f;
        for (int c = 0; c < cnt; ++c) {
            int d = t - s_d[c];
            if (d >= 0) acc += s_a[c] * sg[d];
        }
        op[i * 256 + threadIdx.x] = acc;
    }
}

// ---------------------------------------------------------------------------
extern "C" void kernel_launch(void* const* d_in, const int* in_sizes, int n_in,
                              void* d_out, int out_size, void* d_ws, size_t ws_size,
                              hipStream_t stream) {
    const float* param  = (const float*)d_in[0];
    const float* times  = (const float*)d_in[1];
    const float* noise  = (const float*)d_in[2];
    const float* pos    = (const float*)d_in[3];
    const float* proj_w = (const float*)d_in[4];
    const float* proj_b = (const float*)d_in[5];
    const float* w0 = (const float*)d_in[6];  const float* b0 = (const float*)d_in[7];
    const float* w1 = (const float*)d_in[8];  const float* b1 = (const float*)d_in[9];
    const float* w2 = (const float*)d_in[10]; const float* b2 = (const float*)d_in[11];
    const float* w_out = (const float*)d_in[12]; const float* b_out = (const float*)d_in[13];
    float* out = (float*)d_out;

    float* ws = (float*)d_ws;
    float* x0     = ws;                                  // 32*256
    float* X      = x0 + B_ * CH;                        // 4096*256
    float* H1     = X + (size_t)MROWS * CH;              // 4096*256
    float* H2     = H1 + (size_t)MROWS * CH;             // 4096*256
    float* Y      = H2 + (size_t)MROWS * CH;             // 4096*514
    float* Adft   = Y + (size_t)MROWS * TC;              // 4096*516
    float* Wd     = Adft + (size_t)MROWS * KD;           // 512*516
    float* frames = Wd + (size_t)WIN * KD;               // 4096*512
    float* sig    = frames + (size_t)MROWS * WIN;        // 32*32768

    // independent precompute
    wdft_kernel<<<(WIN * KD + 255) / 256, 256, 0, stream>>>(Wd);

    proj_kernel<<<(B_ * CH + 255) / 256, 256, 0, stream>>>(param, proj_w, proj_b, x0);
    prepx_kernel<<<(MROWS * CH) / 256, 256, 0, stream>>>(x0, pos, X);

    // MLP: three hidden layers (leaky relu) + output layer
    {
        int tiles = (MROWS / 32) * 4;                    // N=256 -> 4 strips
        int blocks = (tiles + 7) / 8;
        gemm_wmma<<<blocks, 256, 0, stream>>>(X,  w0, b0, H1, MROWS, CH, CH, CH, CH, CH, 4, 1);
        gemm_wmma<<<blocks, 256, 0, stream>>>(H1, w1, b1, H2, MROWS, CH, CH, CH, CH, CH, 4, 1);
        gemm_wmma<<<blocks, 256, 0, stream>>>(H2, w2, b2, X,  MROWS, CH, CH, CH, CH, CH, 4, 1);
    }
    {
        int nstrips = (TC + 63) / 64;                    // 9 strips, edge predicated
        int tiles = (MROWS / 32) * nstrips;
        int blocks = (tiles + 7) / 8;
        gemm_wmma<<<blocks, 256, 0, stream>>>(X, w_out, b_out, Y, MROWS, TC, CH, CH, CH, TC, nstrips, 0);
    }

    // phase cumsum -> interleaved re/im spectrum rows
    spec_kernel<<<(B_ * NCOEFFS + 255) / 256, 256, 0, stream>>>(Y, noise, Adft);

    // windowed irfft as GEMM: frames(4096x512) = Adft(4096x516) * Wd^T
    {
        int tiles = (MROWS / 32) * 8;                    // N=512 -> 8 strips
        int blocks = (tiles + 7) / 8;
        gemm_wmma<<<blocks, 256, 0, stream>>>(Adft, Wd, nullptr, frames, MROWS, WIN, KD, KD, KD, WIN, 8, 0);
    }

    // overlap-add
    ola_kernel<<<(B_ * NSAMPLES) / 256, 256, 0, stream>>>(frames, sig);

    // sparse event convolution -> d_out (32,16,32768)
    conv_kernel<<<B_ * NEVENTS * 16, 256, 0, stream>>>(times, sig, out);
}